// NuFFT_42442866819595
// MI455X (gfx1250) — compile-verified
//
#include <hip/hip_runtime.h>
#include <math.h>

typedef __attribute__((ext_vector_type(2))) float v2f;
typedef __attribute__((ext_vector_type(8))) float v8f;

#define NPIX  512
#define NCHAN 4
#define NVIS  131072
#define GRIDN 1024
#define JW    6

// ---- compile-time constants -------------------------------------------------
constexpr double PI_D    = 3.14159265358979323846;
constexpr double DL_D    = 0.005 * PI_D / (180.0 * 3600.0);        // cell in rad
constexpr float  ALPHA_F = (float)(2.34 * 6.0);                    // 14.04
constexpr float  GSCALE  = (float)(1000.0 * DL_D * (double)GRIDN); // klam -> grid units
constexpr float  CELL2   = (float)(0.005 * 0.005);
constexpr float  TWO_PI_F = (float)(2.0 * PI_D);

// ---- CDNA5 async global->LDS copy helpers -----------------------------------
__device__ __forceinline__ void async_g2l_b128(unsigned lds_off, const void* g) {
    // GLOBAL_LOAD_ASYNC_TO_LDS_B128 (GV mode): vdst = LDS byte address,
    // vaddr = 64-bit global address.  Tracked on ASYNCcnt.
    asm volatile("global_load_async_to_lds_b128 %0, %1, off"
                 :: "v"(lds_off), "v"((unsigned long long)(size_t)g)
                 : "memory");
}
__device__ __forceinline__ void wait_async0() {
    asm volatile("s_wait_asynccnt 0x0" ::: "memory");
}

// ---- modified Bessel I0 (Abramowitz & Stegun 9.8.1 / 9.8.2, ~2e-7 rel) ------
__device__ __forceinline__ float besseli0(float x) {
    x = fabsf(x);
    if (x < 3.75f) {
        float t = x / 3.75f; t *= t;
        return 1.0f + t*(3.5156229f + t*(3.0899424f + t*(1.2067492f +
               t*(0.2659732f + t*(0.0360768f + t*0.0045813f)))));
    } else {
        float t = 3.75f / x;
        float p = 0.39894228f + t*(0.01328592f + t*(0.00225319f + t*(-0.00157565f +
                  t*(0.00916281f + t*(-0.02057706f + t*(0.02635537f +
                  t*(-0.01647633f + t*0.00392377f)))))));
        return expf(x) * rsqrtf(x) * p;
    }
}

// gridding-correction 1/kb_ft(n):  s/sinh(s), s = sqrt(alpha^2 - (pi*J*n)^2)
__device__ __forceinline__ float apod(int y) {
    float n = (float)(y - 256) * (1.0f / (float)GRIDN);
    float u = (float)PI_D * (float)JW * n;
    float s = sqrtf(ALPHA_F * ALPHA_F - u * u);   // always > 13.2 here
    return s / sinhf(s);
}

// ---- kernel 1: DFT matrix W[k][n] = exp(-2*pi*i*(k*n mod 1024)/1024) --------
__global__ void k_dftmat(float* __restrict__ wr, float* __restrict__ wi) {
    int t = blockIdx.x * blockDim.x + threadIdx.x;      // 0 .. 1M-1
    int k = t >> 10, n = t & (GRIDN - 1);
    int ph = (k * n) & (GRIDN - 1);                     // exact phase reduction
    float ang = -TWO_PI_F * (float)ph * (1.0f / (float)GRIDN);
    float s, c;
    __sincosf(ang, &s, &c);
    wr[t] = c;
    wi[t] = s;
}

// ---- kernel 2: apodize + fftshift(512) + pad + ifftshift(1024) --------------
__global__ void k_prep(const float* __restrict__ cube, float* __restrict__ P) {
    int t = blockIdx.x * blockDim.x + threadIdx.x;      // 0 .. 4M-1
    int c  = t >> 20;
    int gy = (t >> 10) & (GRIDN - 1);
    int gx = t & (GRIDN - 1);
    int py = (gy + 512) & (GRIDN - 1);                  // undo ifftshift(1024)
    int px = (gx + 512) & (GRIDN - 1);
    float val = 0.0f;
    if (py >= 256 && py < 768 && px >= 256 && px < 768) {
        int y = py - 256, x = px - 256;                 // coords in shifted 512 frame
        int sy = (y + 256) & (NPIX - 1);                // undo fftshift(512)
        int sx = (x + 256) & (NPIX - 1);
        val = cube[((size_t)(c * NPIX + sy)) * NPIX + sx] * apod(y) * apod(x);
    }
    P[t] = val;
}

// ---- kernel 3: complex GEMM via V_WMMA_F32_16X16X4_F32 ----------------------
// C = A * B, all 1024x1024, r/i planes separate.  HASBI=false => B real.
template <bool HASBI>
__global__ __launch_bounds__(256) void k_zgemm(
    const float* __restrict__ Ar, const float* __restrict__ Ai,
    const float* __restrict__ Br, const float* __restrict__ Bi,
    float* __restrict__ Cr, float* __restrict__ Ci)
{
    __shared__ float sAr[16][64];   // A chunk, K-major (transposed via VGPRs)
    __shared__ float sAi[16][64];
    __shared__ float sBr[16][64];   // B chunk, K-major (async global->LDS)
    __shared__ float sBi[16][64];

    const int tid  = threadIdx.x;
    const int bm   = blockIdx.y, bn = blockIdx.x;
    const int lane = tid & 31;
    const int wid  = tid >> 5;              // 8 waves
    const int lr   = lane & 15;
    const int hi   = lane >> 4;             // lane half

    // two 16x16 C tiles per wave (same M row, adjacent N) -> shared A fragments
    const int t0  = wid * 2;
    const int Mb  = ((t0    ) >> 2) * 16;
    const int Nb0 = ((t0    ) & 3) * 16;
    const int Nb1 = ((t0 + 1) & 3) * 16;

    v8f cr0 = {}, ci0 = {}, cr1 = {}, ci1 = {};

    // cooperative-load decomposition: 256 threads x float4
    const int ea  = tid * 4;
    const int am  = ea >> 4;                // A row within 64
    const int ak  = ea & 15;                // A k within 16 (mult of 4)
    const int bk  = ea >> 6;                // B k within 16
    const int bnn = ea & 63;                // B n within 64 (mult of 4)

    const unsigned ldsBr = (unsigned)(size_t)&sBr[bk][bnn];   // low 32 bits = LDS addr
    const unsigned ldsBi = (unsigned)(size_t)&sBi[bk][bnn];

    for (int k0c = 0; k0c < GRIDN; k0c += 16) {
        // 1) kick off async B staging (contiguous copy, no VGPR bounce)
        const size_t boff = (size_t)(k0c + bk) * GRIDN + bn * 64 + bnn;
        async_g2l_b128(ldsBr, Br + boff);
        if (HASBI) async_g2l_b128(ldsBi, Bi + boff);

        // 2) A staging with transpose (overlaps with the async copies)
        {
            const size_t aoff = (size_t)(bm * 64 + am) * GRIDN + k0c + ak;
            float4 va = *(const float4*)(Ar + aoff);
            sAr[ak + 0][am] = va.x; sAr[ak + 1][am] = va.y;
            sAr[ak + 2][am] = va.z; sAr[ak + 3][am] = va.w;
            float4 vi = *(const float4*)(Ai + aoff);
            sAi[ak + 0][am] = vi.x; sAi[ak + 1][am] = vi.y;
            sAi[ak + 2][am] = vi.z; sAi[ak + 3][am] = vi.w;
        }
        wait_async0();
        __syncthreads();

        #pragma unroll
        for (int kk = 0; kk < 16; kk += 4) {
            const int ka = kk + 2 * hi;     // fragment K selection per lane half
            v2f ar, ai, br0, br1;
            ar.x  = sAr[ka][Mb  + lr]; ar.y  = sAr[ka + 1][Mb  + lr];
            ai.x  = sAi[ka][Mb  + lr]; ai.y  = sAi[ka + 1][Mb  + lr];
            br0.x = sBr[ka][Nb0 + lr]; br0.y = sBr[ka + 1][Nb0 + lr];
            br1.x = sBr[ka][Nb1 + lr]; br1.y = sBr[ka + 1][Nb1 + lr];

            cr0 = __builtin_amdgcn_wmma_f32_16x16x4_f32(false, ar, false, br0, (short)0, cr0, false, false);
            ci0 = __builtin_amdgcn_wmma_f32_16x16x4_f32(false, ai, false, br0, (short)0, ci0, false, false);
            cr1 = __builtin_amdgcn_wmma_f32_16x16x4_f32(false, ar, false, br1, (short)0, cr1, false, false);
            ci1 = __builtin_amdgcn_wmma_f32_16x16x4_f32(false, ai, false, br1, (short)0, ci1, false, false);

            if (HASBI) {
                v2f bi0, bi1, na;
                bi0.x = sBi[ka][Nb0 + lr]; bi0.y = sBi[ka + 1][Nb0 + lr];
                bi1.x = sBi[ka][Nb1 + lr]; bi1.y = sBi[ka + 1][Nb1 + lr];
                na.x = -ai.x; na.y = -ai.y;        // f32 WMMA has no A-neg: pre-negate
                cr0 = __builtin_amdgcn_wmma_f32_16x16x4_f32(false, na, false, bi0, (short)0, cr0, false, false);
                ci0 = __builtin_amdgcn_wmma_f32_16x16x4_f32(false, ar, false, bi0, (short)0, ci0, false, false);
                cr1 = __builtin_amdgcn_wmma_f32_16x16x4_f32(false, na, false, bi1, (short)0, cr1, false, false);
                ci1 = __builtin_amdgcn_wmma_f32_16x16x4_f32(false, ar, false, bi1, (short)0, ci1, false, false);
            }
        }
        __syncthreads();
    }

    // C/D layout: VGPR v -> rows v (lanes 0-15) and v+8 (lanes 16-31)
    #pragma unroll
    for (int v = 0; v < 8; ++v) {
        int r  = bm * 64 + Mb + v + 8 * hi;
        int c0 = bn * 64 + Nb0 + lr;
        int c1 = bn * 64 + Nb1 + lr;
        Cr[(size_t)r * GRIDN + c0] = cr0[v];
        Ci[(size_t)r * GRIDN + c0] = ci0[v];
        Cr[(size_t)r * GRIDN + c1] = cr1[v];
        Ci[(size_t)r * GRIDN + c1] = ci1[v];
    }
}

// ---- kernel 4: per-visibility Kaiser-Bessel weights & indices ---------------
__global__ void k_weights(const float* __restrict__ uu, const float* __restrict__ vv,
                          float* __restrict__ wU, float* __restrict__ wV,
                          int* __restrict__ iU, int* __restrict__ iV)
{
    int m = blockIdx.x * blockDim.x + threadIdx.x;
    if (m >= NVIS) return;
    float gu = uu[m] * GSCALE;
    float gv = vv[m] * GSCALE;
    float bu = floorf(gu), bv = floorf(gv);
    #pragma unroll
    for (int t = 0; t < JW; ++t) {
        float pu = bu + (float)(t - 2);
        float pv = bv + (float)(t - 2);
        float du = (gu - pu) * (1.0f / 3.0f);      // 2d/J
        float dv = (gv - pv) * (1.0f / 3.0f);
        float tu = 1.0f - du * du;
        float tv = 1.0f - dv * dv;
        wU[m * JW + t] = (tu > 0.0f) ? besseli0(ALPHA_F * sqrtf(tu)) * (1.0f / 6.0f) : 0.0f;
        wV[m * JW + t] = (tv > 0.0f) ? besseli0(ALPHA_F * sqrtf(tv)) * (1.0f / 6.0f) : 0.0f;
        iU[m * JW + t] = ((int)pu) & (GRIDN - 1);  // two's-complement & == mod 1024
        iV[m * JW + t] = ((int)pv) & (GRIDN - 1);
    }
}

// ---- kernel 5: 6x6 gather + separable weighting (L2-resident) ---------------
__global__ void k_gather(const float* __restrict__ Fr, const float* __restrict__ Fi,
                         const float* __restrict__ wU, const float* __restrict__ wV,
                         const int* __restrict__ iU, const int* __restrict__ iV,
                         float* __restrict__ out)
{
    int m = blockIdx.x * blockDim.x + threadIdx.x;
    int c = blockIdx.y;
    const float* fr = Fr + (size_t)c * GRIDN * GRIDN;
    const float* fi = Fi + (size_t)c * GRIDN * GRIDN;

    float lu[JW], lv[JW];
    int   ju[JW], jv[JW];
    #pragma unroll
    for (int t = 0; t < JW; ++t) {
        lu[t] = wU[m * JW + t]; lv[t] = wV[m * JW + t];
        ju[t] = iU[m * JW + t]; jv[t] = iV[m * JW + t];
    }
    float re = 0.0f, im = 0.0f;
    #pragma unroll
    for (int i = 0; i < JW; ++i) {
        size_t row = (size_t)jv[i] * GRIDN;
        float pr = 0.0f, pi = 0.0f;
        #pragma unroll
        for (int j = 0; j < JW; ++j) {
            size_t a = row + ju[j];
            pr += fr[a] * lu[j];
            pi += fi[a] * lu[j];
        }
        re += lv[i] * pr;
        im += lv[i] * pi;
    }
    size_t o = ((size_t)c * NVIS + m) * 2;
    out[o + 0] = re * CELL2;
    out[o + 1] = im * CELL2;
}

// ---- host launch ------------------------------------------------------------
extern "C" void kernel_launch(void* const* d_in, const int* in_sizes, int n_in,
                              void* d_out, int out_size, void* d_ws, size_t ws_size,
                              hipStream_t stream) {
    (void)in_sizes; (void)n_in; (void)out_size; (void)ws_size;
    const float* cube = (const float*)d_in[0];
    const float* uu   = (const float*)d_in[1];
    const float* vv   = (const float*)d_in[2];

    constexpr size_t M1 = (size_t)GRIDN * GRIDN;   // 1M floats
    float* ws  = (float*)d_ws;
    float* Wr  = ws;                 // 1M
    float* Wi  = ws + 1 * M1;        // 1M
    float* P   = ws + 2 * M1;        // 4M (4 channels)
    float* F1r = ws + 6 * M1;        // 4M
    float* F1i = ws + 10 * M1;       // 4M
    float* Fr  = ws + 14 * M1;       // 4M
    float* Fi  = ws + 18 * M1;       // 4M
    float* wU  = ws + 22 * M1;
    float* wV  = wU + (size_t)NVIS * JW;
    int*   iU  = (int*)(wV + (size_t)NVIS * JW);
    int*   iV  = iU + (size_t)NVIS * JW;

    k_dftmat<<<(int)(M1 / 256), 256, 0, stream>>>(Wr, Wi);
    k_prep<<<(int)(NCHAN * M1 / 256), 256, 0, stream>>>(cube, P);

    dim3 gg(GRIDN / 64, GRIDN / 64);
    for (int c = 0; c < NCHAN; ++c)   // column DFT: F1 = W * P  (B real)
        k_zgemm<false><<<gg, 256, 0, stream>>>(Wr, Wi, P + c * M1, nullptr,
                                               F1r + c * M1, F1i + c * M1);
    for (int c = 0; c < NCHAN; ++c)   // row DFT: F = F1 * W  (full complex)
        k_zgemm<true><<<gg, 256, 0, stream>>>(F1r + c * M1, F1i + c * M1, Wr, Wi,
                                              Fr + c * M1, Fi + c * M1);

    k_weights<<<NVIS / 256, 256, 0, stream>>>(uu, vv, wU, wV, iU, iV);
    dim3 ga(NVIS / 256, NCHAN);
    k_gather<<<ga, 256, 0, stream>>>(Fr, Fi, wU, wV, iU, iV, (float*)d_out);
}